// PhotoreceptorLayer_39711267619024
// MI455X (gfx1250) — compile-verified
//
#include <hip/hip_runtime.h>
#include <stdint.h>
#include <math.h>

// Photoreceptor (Rieke) scan for MI455X / gfx1250.
// B=16, U=1, T=1000, D=2500. 40,000 independent channels, 1000 serial steps.
// HBM floor: 320 MB / 23.3 TB/s ~= 14 us. Input staged via Tensor Data Mover
// (tensor_load_to_lds, TENSORcnt) into double-buffered LDS; fast path for
// hill==3 / hc==4 has zero libm calls and one v_rcp per step.

#define BW   128   // channels (threads) per block
#define ROWS 64    // time rows per TDM chunk; 2*ROWS*BW*4 = 64 KB LDS

constexpr int kT = 1000;
constexpr int kD = 2500;
constexpr int kB = 16;

typedef unsigned int v4u __attribute__((ext_vector_type(4)));
typedef int          v8i __attribute__((ext_vector_type(8)));
typedef int          v4i __attribute__((ext_vector_type(4)));

__device__ __forceinline__ float fast_pow(float x, float h) {
  // x > 0 on all paths that reach here
  return __builtin_amdgcn_exp2f(h * __builtin_amdgcn_logf(x));
}

// Issue one TDM 2D tile load: tile_rows x BW floats, row stride = `stride`
// elements, from gsrc into LDS byte offset lds_byte. Dims are "remaining
// extent from tile start" so out-of-range columns zero-fill (D# OOB rule).
__device__ __forceinline__ void tdm_tile_load(const float* gsrc, uint32_t lds_byte,
                                              uint32_t cols_left, uint32_t rows_left,
                                              uint32_t tile_rows, uint32_t stride) {
  uint64_t ga = (uint64_t)(uintptr_t)gsrc;
  v4u g0;
  g0[0] = 1u;                                   // count=1, user descriptor
  g0[1] = lds_byte;                             // lds_addr [63:32]
  g0[2] = (uint32_t)ga;                         // global_addr [95:64]
  g0[3] = (uint32_t)((ga >> 32) & 0x01FFFFFFu)  // global_addr [120:96]
        | (2u << 30);                           // type=2 ("image")
  v8i g1;
  g1[0] = (int)(2u << 16);                      // workgroup_mask=0, data_size=4B
  g1[1] = (int)((cols_left & 0xFFFFu) << 16);   // tensor_dim0 lo16 @bit48
  g1[2] = (int)((cols_left >> 16) | ((rows_left & 0xFFFFu) << 16)); // dim0 hi / dim1 lo
  g1[3] = (int)((rows_left >> 16) | ((uint32_t)BW << 16));          // dim1 hi / tile_dim0
  g1[4] = (int)(tile_rows & 0xFFFFu);           // tile_dim1; tile_dim2=0
  g1[5] = (int)stride;                          // tensor_dim0_stride lo32
  g1[6] = 0;                                    // stride hi16 / dim1_stride lo16
  g1[7] = 0;
  v4i gz;  gz[0] = 0; gz[1] = 0; gz[2] = 0; gz[3] = 0;
  v8i gz8; gz8[0] = 0; gz8[1] = 0; gz8[2] = 0; gz8[3] = 0;
  gz8[4] = 0; gz8[5] = 0; gz8[6] = 0; gz8[7] = 0;
  // 6-arg form (clang-23 / therock-10.0 headers)
  __builtin_amdgcn_tensor_load_to_lds(g0, g1, gz, gz, gz8, 0);
}

struct PK {
  float dt, sig, ph, et, gam, c2c, hill, hc, inv_ha, cd;
  float smax, gd2, k_r, k_p, k_c, d_et, Agh, nc2;
};

template <bool F3, bool F4>
__device__ void scan_loop(const PK pk, const float* __restrict__ xb,
                          float* __restrict__ ob, float (*lbuf)[ROWS * BW],
                          int d0, bool valid) {
  const int tid = threadIdx.x;
  constexpr int nrows = kT - 1;                   // x rows 0..T-2 are consumed
  constexpr int nchunk = (nrows + ROWS - 1) / ROWS;
  const uint32_t l0 = (uint32_t)(uintptr_t)&lbuf[0][0];
  const uint32_t l1 = (uint32_t)(uintptr_t)&lbuf[1][0];

  if (tid < 32) {                                 // wave 0 drives the TDM
    constexpr int rows0 = nrows < ROWS ? nrows : ROWS;
    tdm_tile_load(xb, l0, (uint32_t)(kD - d0), (uint32_t)kT,
                  (uint32_t)rows0, (uint32_t)kD);
    __builtin_amdgcn_s_wait_tensorcnt(0);
  }
  __syncthreads();

  // Initial state from x row 0
  const float x0 = lbuf[0][tid];
  float r = x0 * pk.gam / pk.sig;
  float p = (pk.et + r) / pk.ph;
  float c = pk.cd;
  float s = pk.gd2 * pk.et / pk.ph;
  float g = pk.gd2;
  float gh = powf(g, pk.hill);                    // carried g^hill

  // out row 0: g_full[0]=0 -> -(c2c*0^hill)/2 ; 0^hill = 0 for hill>0, 1 for hill==0
  if (valid) ob[0] = (pk.hill == 0.0f) ? pk.nc2 : pk.nc2 * 0.0f;

  const float ndt = -pk.dt;
  int buf = 0;
#pragma unroll 1
  for (int ci = 0; ci < nchunk; ++ci) {
    const int t0 = ci * ROWS;
    const int rows = (nrows - t0) < ROWS ? (nrows - t0) : ROWS;

    if (ci + 1 < nchunk && tid < 32) {            // prefetch next chunk
      const int t0n = t0 + ROWS;
      const int rowsn = (nrows - t0n) < ROWS ? (nrows - t0n) : ROWS;
      tdm_tile_load(xb + (size_t)t0n * kD, buf ? l0 : l1,
                    (uint32_t)(kD - d0), (uint32_t)(kT - t0n),
                    (uint32_t)rowsn, (uint32_t)kD);
    }

    const float* lb = &lbuf[buf][0];
    float* op = ob + (size_t)(t0 + 1) * kD;
#pragma unroll 4
    for (int rr = 0; rr < rows; ++rr) {
      const float xv = lb[rr * BW + tid];
      const float r1 = fmaf(pk.k_r, r, pk.gam * xv);
      const float p1 = fmaf(pk.k_p, p, fmaf(pk.dt, r, pk.d_et));
      const float c1 = fmaf(pk.k_c, c, pk.Agh * gh);
      const float u  = c1 * pk.inv_ha;
      const float u2 = u * u;
      const float up = F4 ? u2 * u2 : fast_pow(u, pk.hc);
      const float s1 = pk.smax * __builtin_amdgcn_rcpf(1.0f + up);
      const float g1 = fmaf(ndt * p, g, fmaf(pk.dt, s, g));
      const float gh1 = F3 ? g1 * g1 * g1 : fast_pow(g1, pk.hill);
      if (valid) __builtin_nontemporal_store(pk.nc2 * gh1, &op[(size_t)rr * kD]);
      r = r1; p = p1; c = c1; s = s1; g = g1; gh = gh1;
    }

    __syncthreads();                              // all waves done with lbuf[buf]
    if (ci + 1 < nchunk) {
      if (tid < 32) __builtin_amdgcn_s_wait_tensorcnt(0);
      __syncthreads();                            // next chunk's DMA complete
    }
    buf ^= 1;
  }
}

__global__ __launch_bounds__(BW) void photoreceptor_kernel(
    const float* __restrict__ x,
    const float* sigma_, const float* phi_, const float* eta_, const float* beta_,
    const float* c2c_, const float* hill_, const float* cd_,
    const float* hc_, const float* ha_, const float* gamma_, const float* gdark_,
    const float* timeBin_, float* __restrict__ out) {
  __shared__ float lbuf[2][ROWS * BW];
  constexpr int bpb = (kD + BW - 1) / BW;         // 20, compile-time
  const int bb = blockIdx.x / bpb;
  const int d0 = (blockIdx.x % bpb) * BW;
  const int d  = d0 + threadIdx.x;
  const bool valid = d < kD;

  const float timeBin = *timeBin_;
  PK pk;
  pk.dt  = 0.001f * timeBin;
  pk.sig = *sigma_ * 10.0f;
  pk.ph  = *phi_ * 10.0f;
  pk.et  = *eta_ * 1000.0f;
  const float be = *beta_ * 10.0f;
  pk.gam = *gamma_ * 10.0f / timeBin;
  const float gd = *gdark_ * 100.0f;
  pk.c2c  = *c2c_;
  pk.hill = *hill_;
  pk.hc   = *hc_;
  const float ha = *ha_;
  pk.inv_ha = 1.0f / ha;
  pk.cd = *cd_;

  const float darkCurrent = powf(gd, pk.hill) * pk.c2c * 0.5f;
  pk.gd2 = powf(2.0f * darkCurrent / pk.c2c, 1.0f / pk.hill);
  const float cur2ca = be * pk.cd / darkCurrent;
  pk.smax = pk.et / pk.ph * pk.gd2 * (1.0f + powf(pk.cd / ha, pk.hc));
  pk.k_r  = 1.0f - pk.dt * pk.sig;
  pk.k_p  = 1.0f - pk.dt * pk.ph;
  pk.k_c  = 1.0f - pk.dt * be;
  pk.d_et = pk.dt * pk.et;
  pk.Agh  = pk.dt * cur2ca * pk.c2c * 0.5f;
  pk.nc2  = -0.5f * pk.c2c;

  const float* xb = x   + (size_t)bb * (size_t)kT * (size_t)kD + d0;
  float*       ob = out + (size_t)bb * (size_t)kT * (size_t)kD + d;

  // Uniform runtime dispatch: exact reference exponents get the mul-only path.
  if (pk.hill == 3.0f && pk.hc == 4.0f)
    scan_loop<true, true>(pk, xb, ob, lbuf, d0, valid);
  else
    scan_loop<false, false>(pk, xb, ob, lbuf, d0, valid);
}

extern "C" void kernel_launch(void* const* d_in, const int* in_sizes, int n_in,
                              void* d_out, int out_size, void* d_ws, size_t ws_size,
                              hipStream_t stream) {
  const float* x      = (const float*)d_in[0];
  const float* sigma_ = (const float*)d_in[1];
  const float* phi_   = (const float*)d_in[2];
  const float* eta_   = (const float*)d_in[3];
  const float* beta_  = (const float*)d_in[4];
  const float* c2c_   = (const float*)d_in[5];
  const float* hill_  = (const float*)d_in[6];
  const float* cd_    = (const float*)d_in[7];
  // d_in[8] = betaSlow (unused by the reference model)
  const float* hc_    = (const float*)d_in[9];
  const float* ha_    = (const float*)d_in[10];
  const float* gamma_ = (const float*)d_in[11];
  const float* gdark_ = (const float*)d_in[12];
  const float* tb_    = (const float*)d_in[13];
  float* out = (float*)d_out;

  const int bpb = (kD + BW - 1) / BW;    // 20 blocks per batch row
  dim3 grid(kB * bpb), block(BW);        // 320 blocks x 128 threads
  hipLaunchKernelGGL(photoreceptor_kernel, grid, block, 0, stream,
                     x, sigma_, phi_, eta_, beta_, c2c_, hill_, cd_,
                     hc_, ha_, gamma_, gdark_, tb_, out);
}